// MatchModule_24979529794229
// MI455X (gfx1250) — compile-verified
//
#include <hip/hip_runtime.h>

// ---------------- problem constants (match reference module) ----------------
constexpr int Bn    = 32;            // batches
constexpr int PPB   = 256;           // proposals per batch
constexpr int Np    = Bn * PPB;      // 8192 proposals
constexpr int Mf    = 128;           // vote feature dim
constexpr int LANGD = 256;           // language dim
constexpr int Hd    = 128;           // hidden dim
constexpr int Kfuse = LANGD + Mf;    // 384
#define BN_EPS 1e-5f

typedef float v2f __attribute__((ext_vector_type(2)));
typedef float v8f __attribute__((ext_vector_type(8)));

__device__ __forceinline__ v8f wmma4(v2f a, v2f b, v8f c) {
  // D(16x16,f32) = A(16x4,f32) * B(4x16,f32) + C   -> v_wmma_f32_16x16x4_f32
  return __builtin_amdgcn_wmma_f32_16x16x4_f32(false, a, false, b, (short)0, c,
                                               false, false);
}

#define V8F_ZERO {0.f, 0.f, 0.f, 0.f, 0.f, 0.f, 0.f, 0.f}

// ---------------- scatter preamble -----------------------------------------
// bid[i] = searchsorted(b_offsets, first_pt_i, 'right') - 1
__global__ void k_bid(const int* __restrict__ pidx, const int* __restrict__ poff,
                      const int* __restrict__ boff, int* __restrict__ bid) {
  int i = blockIdx.x * blockDim.x + threadIdx.x;
  if (i >= Np) return;
  int first_pt = pidx[2 * poff[i] + 1];   // proposals_idx[offset, 1]
  int b = -1;
#pragma unroll
  for (int j = 0; j <= Bn; ++j) b += (boff[j] <= first_pt) ? 1 : 0;
  bid[i] = b;
}

// stable rank within bucket: rank_i = #{ j < i : bid[j] == bid[i] }
__global__ __launch_bounds__(256) void k_rank(const int* __restrict__ bid,
                                              int* __restrict__ dst) {
  __shared__ int sbid[Np];                // 32 KB LDS
  for (int j = threadIdx.x; j < Np; j += blockDim.x) sbid[j] = bid[j];
  __syncthreads();
  int i = blockIdx.x * blockDim.x + threadIdx.x;
  if (i >= Np) return;
  int myb = sbid[i];
  int r = 0;
  for (int j = 0; j < i; ++j) r += (sbid[j] == myb) ? 1 : 0;
  dst[i] = myb * PPB + r;
}

__global__ void k_zero(float* __restrict__ p, int n) {
  int i = blockIdx.x * blockDim.x + threadIdx.x;
  if (i < n) p[i] = 0.f;
}

// padded[dst[i], :] = feats[i, :]
__global__ __launch_bounds__(128) void k_scatter(const float* __restrict__ feats,
                                                 const int* __restrict__ dst,
                                                 float* __restrict__ padded) {
  int i = blockIdx.x;
  int d = dst[i];
  if (d < 0 || d >= Np) return;
  padded[(size_t)d * Mf + threadIdx.x] = feats[(size_t)i * Mf + threadIdx.x];
}

// ---------------- Layer 1: x1 = relu(W_fuse @ [padded|lang]^T + b_fuse) ----
// Grid (Bn, 4): one batch b, 64-wide p slice per WG. 8 waves = 8 o-tiles.
// Each wave: o-row (16 outs) x 4 p-tiles; A-frag shared by 4 WMMAs; lang-half
// B-frag is column-broadcast -> shared by all 4 p-tiles as well.
__global__ __launch_bounds__(256) void k_l1(const float* __restrict__ padded,
                                            const float* __restrict__ lang,
                                            const float* __restrict__ Wf,
                                            const float* __restrict__ bf,
                                            float* __restrict__ out) {
  const int b     = blockIdx.x;
  const int pbase = blockIdx.y * 64;
  const int wave  = threadIdx.x >> 5;        // 0..7 -> o-tile
  const int lane  = threadIdx.x & 31;
  const int half  = lane >> 4;
  const int lm    = lane & 15;
  const int o0    = wave << 4;

  const float* arow = Wf + (size_t)(o0 + lm) * Kfuse + 2 * half;  // k = k0 (+1)
  const float* xb   = padded + (size_t)b * PPB * Mf;
  const float* br0  = xb + (size_t)(pbase +  0 + lm) * Mf + 2 * half;
  const float* br1  = xb + (size_t)(pbase + 16 + lm) * Mf + 2 * half;
  const float* br2  = xb + (size_t)(pbase + 32 + lm) * Mf + 2 * half;
  const float* br3  = xb + (size_t)(pbase + 48 + lm) * Mf + 2 * half;
  const float* lb   = lang + (size_t)b * LANGD + 2 * half;        // k-Mf = k0 (+1)

  v8f acc0 = V8F_ZERO, acc1 = V8F_ZERO, acc2 = V8F_ZERO, acc3 = V8F_ZERO;

  // ---- K segment 1: c in [0,128) from padded rows -------------------------
#pragma unroll 4
  for (int k0 = 0; k0 < Mf; k0 += 4) {
    v2f a;   a.x  = arow[k0];      a.y  = arow[k0 + 1];
    v2f b0;  b0.x = br0[k0];       b0.y = br0[k0 + 1];
    v2f b1;  b1.x = br1[k0];       b1.y = br1[k0 + 1];
    v2f b2;  b2.x = br2[k0];       b2.y = br2[k0 + 1];
    v2f b3;  b3.x = br3[k0];       b3.y = br3[k0 + 1];
    acc0 = wmma4(a, b0, acc0);
    acc1 = wmma4(a, b1, acc1);
    acc2 = wmma4(a, b2, acc2);
    acc3 = wmma4(a, b3, acc3);
  }
  // ---- K segment 2: c in [128,384) from lang (broadcast over p) -----------
#pragma unroll 4
  for (int k0 = 0; k0 < LANGD; k0 += 4) {
    v2f a;   a.x = arow[Mf + k0];  a.y = arow[Mf + k0 + 1];
    v2f bb;  bb.x = lb[k0];        bb.y = lb[k0 + 1];
    acc0 = wmma4(a, bb, acc0);
    acc1 = wmma4(a, bb, acc1);
    acc2 = wmma4(a, bb, acc2);
    acc3 = wmma4(a, bb, acc3);
  }

  float* ob = out + ((size_t)b * Hd) * PPB + pbase + lm;
#pragma unroll
  for (int v = 0; v < 8; ++v) {
    const int o = o0 + v + 8 * half;
    const float bias = bf[o];
    float* orow = ob + (size_t)o * PPB;
    orow[ 0] = fmaxf(acc0[v] + bias, 0.f);
    orow[16] = fmaxf(acc1[v] + bias, 0.f);
    orow[32] = fmaxf(acc2[v] + bias, 0.f);
    orow[48] = fmaxf(acc3[v] + bias, 0.f);
  }
}

// ---------------- Layers 2/3: out = BN(relu(W @ in + bias)) ----------------
__global__ __launch_bounds__(256) void k_mid(const float* __restrict__ in,
                                             const float* __restrict__ W,
                                             const float* __restrict__ bias,
                                             const float* __restrict__ gam,
                                             const float* __restrict__ bet,
                                             const float* __restrict__ rme,
                                             const float* __restrict__ rva,
                                             float* __restrict__ out) {
  const int b     = blockIdx.x;
  const int pbase = blockIdx.y * 64;
  const int wave  = threadIdx.x >> 5;        // 0..7 -> o-tile
  const int lane  = threadIdx.x & 31;
  const int half  = lane >> 4;
  const int lm    = lane & 15;
  const int o0    = wave << 4;

  const float* arow = W + (size_t)(o0 + lm) * Hd + 2 * half;     // k = k0 (+1)
  const float* inb  = in + (size_t)b * Hd * PPB;
  const float* bc   = inb + (size_t)(2 * half) * PPB + pbase + lm; // row k0, col lm

  v8f acc0 = V8F_ZERO, acc1 = V8F_ZERO, acc2 = V8F_ZERO, acc3 = V8F_ZERO;

#pragma unroll 4
  for (int k0 = 0; k0 < Hd; k0 += 4) {
    const int r0 = k0 * PPB;            // row k0+2*half   (bc already offset)
    const int r1 = r0 + PPB;            // row k0+2*half+1
    v2f a;   a.x  = arow[k0];      a.y  = arow[k0 + 1];
    v2f b0;  b0.x = bc[r0 +  0];   b0.y = bc[r1 +  0];
    v2f b1;  b1.x = bc[r0 + 16];   b1.y = bc[r1 + 16];
    v2f b2;  b2.x = bc[r0 + 32];   b2.y = bc[r1 + 32];
    v2f b3;  b3.x = bc[r0 + 48];   b3.y = bc[r1 + 48];
    acc0 = wmma4(a, b0, acc0);
    acc1 = wmma4(a, b1, acc1);
    acc2 = wmma4(a, b2, acc2);
    acc3 = wmma4(a, b3, acc3);
  }

  float* ob = out + ((size_t)b * Hd) * PPB + pbase + lm;
#pragma unroll
  for (int v = 0; v < 8; ++v) {
    const int o = o0 + v + 8 * half;
    const float bs    = bias[o];
    const float scale = gam[o] * rsqrtf(rva[o] + BN_EPS);
    const float mu    = rme[o];
    const float beta  = bet[o];
    float* orow = ob + (size_t)o * PPB;
    orow[ 0] = (fmaxf(acc0[v] + bs, 0.f) - mu) * scale + beta;
    orow[16] = (fmaxf(acc1[v] + bs, 0.f) - mu) * scale + beta;
    orow[32] = (fmaxf(acc2[v] + bs, 0.f) - mu) * scale + beta;
    orow[48] = (fmaxf(acc3[v] + bs, 0.f) - mu) * scale + beta;
  }
}

// ---------------- Final: conf[b,p] = W3 . x3[b,:,p] + b3 -------------------
__global__ __launch_bounds__(256) void k_final(const float* __restrict__ in,
                                               const float* __restrict__ W3,
                                               const float* __restrict__ b3,
                                               float* __restrict__ out) {
  int idx = blockIdx.x * blockDim.x + threadIdx.x;
  if (idx >= Bn * PPB) return;
  int b = idx / PPB, p = idx % PPB;
  const float* inb = in + (size_t)b * Hd * PPB + p;
  float s = b3[0];
#pragma unroll 8
  for (int c = 0; c < Hd; ++c) s += W3[c] * inb[(size_t)c * PPB];
  out[idx] = s;
}

// ---------------- host side ------------------------------------------------
extern "C" void kernel_launch(void* const* d_in, const int* in_sizes, int n_in,
                              void* d_out, int out_size, void* d_ws, size_t ws_size,
                              hipStream_t stream) {
  const float* feats = (const float*)d_in[0];
  const int*   pidx  = (const int*)d_in[1];
  const int*   poff  = (const int*)d_in[2];
  const int*   boff  = (const int*)d_in[3];
  const float* lang  = (const float*)d_in[4];
  const float* Wf    = (const float*)d_in[5];
  const float* bf    = (const float*)d_in[6];
  const float* W1p   = (const float*)d_in[7];
  const float* b1p   = (const float*)d_in[8];
  const float* g1p   = (const float*)d_in[9];
  const float* be1p  = (const float*)d_in[10];
  const float* rm1p  = (const float*)d_in[11];
  const float* rv1p  = (const float*)d_in[12];
  const float* W2p   = (const float*)d_in[13];
  const float* b2p   = (const float*)d_in[14];
  const float* g2p   = (const float*)d_in[15];
  const float* be2p  = (const float*)d_in[16];
  const float* rm2p  = (const float*)d_in[17];
  const float* rv2p  = (const float*)d_in[18];
  const float* W3p   = (const float*)d_in[19];
  const float* b3p   = (const float*)d_in[20];

  char* ws = (char*)d_ws;
  int*   bid    = (int*)ws;                                    // 32 KB
  int*   dst    = (int*)(ws + (size_t)Np * sizeof(int));       // 32 KB
  float* padded = (float*)(ws + 2 * (size_t)Np * sizeof(int)); // 4 MB
  float* act1   = padded + (size_t)Np * Mf;                    // 4 MB
  float* act2   = padded;                  // alias: padded dead after L1

  k_bid <<<Np / 256, 256, 0, stream>>>(pidx, poff, boff, bid);
  k_rank<<<Np / 256, 256, 0, stream>>>(bid, dst);
  k_zero<<<(Np * Mf) / 256, 256, 0, stream>>>(padded, Np * Mf);
  k_scatter<<<Np, 128, 0, stream>>>(feats, dst, padded);

  dim3 grid(Bn, 4);  // 32 batches x 4 p-slices of 64 = 128 WGs, 8 waves each
  k_l1 <<<grid, 256, 0, stream>>>(padded, lang, Wf, bf, act1);
  k_mid<<<grid, 256, 0, stream>>>(act1, W1p, b1p, g1p, be1p, rm1p, rv1p, act2);
  k_mid<<<grid, 256, 0, stream>>>(act2, W2p, b2p, g2p, be2p, rm2p, rv2p, act1);
  k_final<<<Np / 256, 256, 0, stream>>>(act1, W3p, b3p, (float*)d_out);
}